// Attention_37890201485423
// MI455X (gfx1250) — compile-verified
//
#include <hip/hip_runtime.h>
#include <math.h>

typedef float v2f __attribute__((ext_vector_type(2)));
typedef float v4f __attribute__((ext_vector_type(4)));
typedef float v8f __attribute__((ext_vector_type(8)));

#define DIM    5120
#define NH     40
#define NKV    8
#define HD     128
#define NREP   5          // NH / NKV
#define BS     8
#define KLEN   4096       // start_pos + seqlen
#define QKVC   7168       // 5120 (q) + 1024 (k) + 1024 (v)
#define CH     256        // keys per attention chunk
#define NCHUNK 16         // KLEN / CH
#define KSPLIT 4          // K-dim split for the projection GEMVs

// ---- workspace layout (float offsets) ----
#define WS_Y   0                                    // [BS][QKVC] fused qkv
#define WS_ATT (WS_Y  + BS*QKVC)                    // [BS][DIM]
#define WS_PM  (WS_ATT + BS*DIM)                    // [BS][NKV][NCHUNK][NREP]
#define WS_PL  (WS_PM + BS*NKV*NCHUNK*NREP)
#define WS_PO  (WS_PL + BS*NKV*NCHUNK*NREP)         // [..][NREP][HD]
#define WS_QP  (WS_PO + BS*NKV*NCHUNK*NREP*HD)      // [KSPLIT][BS][QKVC]
#define WS_OP  (WS_QP + KSPLIT*BS*QKVC)             // [KSPLIT][BS][DIM]

// ============================================================
// K-slice of a 16-column GEMV tile via f32 WMMA.
// STRIDE is compile-time so weight loads become base + imm24 offsets.
// ldsX rows 8..15 are pre-zeroed by caller -> A read is branch-free.
// ============================================================
template<int STRIDE>
__device__ __forceinline__ void gemv_ktile(const float* __restrict__ W,
                                           const float* __restrict__ X,
                                           int cl, int dt0, int dt1,
                                           int lane, v8f& acc, float* ldsX)
{
    const int row  = lane & 15;          // A: M row / B: N column
    const int offs = (lane >> 4) << 1;   // K sub-offset 0 or 2
    const float* wp = W + ((size_t)dt0*128 + offs) * STRIDE + cl + row;

    for (int dt = dt0; dt < dt1; ++dt) {
        // stage X tile [8][128] into LDS (padded stride 132)
        for (int i = 0; i < 8; ++i) {
            int f4 = i*32 + lane;
            int bb = f4 >> 5, pp = f4 & 31;
            *(v4f*)&ldsX[bb*132 + pp*4] =
                *(const v4f*)&X[(size_t)bb*DIM + (size_t)dt*128 + pp*4];
        }
        __syncthreads();
#pragma unroll
        for (int t = 0; t < 32; ++t) {
            v2f a = *(const v2f*)&ldsX[row*132 + 4*t + offs];
            v2f b;
            b.x = wp[0];
            b.y = wp[STRIDE];
            acc = __builtin_amdgcn_wmma_f32_16x16x4_f32(false, a, false, b,
                                                        (short)0, acc, false, false);
            wp += (size_t)4 * STRIDE;
        }
        __syncthreads();
    }
}

// ============================================================
// Fused QKV projection partials: qp[ks][b][0:7168]
// grid = (QKVC/16, KSPLIT), one wave per block.
// ============================================================
__global__ __launch_bounds__(32)
void qkv_gemv(const float* __restrict__ x,
              const float* __restrict__ wq,
              const float* __restrict__ wk,
              const float* __restrict__ wv,
              float* __restrict__ qp)
{
    __shared__ float ldsX[16 * 132];
    const int lane = threadIdx.x;
    for (int j = lane; j < 8*132; j += 32) ldsX[8*132 + j] = 0.f;  // zero pad rows
    __syncthreads();

    const int col0 = blockIdx.x * 16;
    const int ks   = blockIdx.y;
    const int dt0  = ks * (DIM/128/KSPLIT);
    const int dt1  = dt0 + (DIM/128/KSPLIT);

    v8f acc = {0.f,0.f,0.f,0.f,0.f,0.f,0.f,0.f};
    if (col0 < DIM)
        gemv_ktile<DIM >(wq, x, col0,            dt0, dt1, lane, acc, ldsX);
    else if (col0 < DIM + 1024)
        gemv_ktile<1024>(wk, x, col0 - DIM,      dt0, dt1, lane, acc, ldsX);
    else
        gemv_ktile<1024>(wv, x, col0 - DIM-1024, dt0, dt1, lane, acc, ldsX);

    if (lane < 16) {
        float* dst = qp + (size_t)ks * BS * QKVC;
        for (int j = 0; j < 8; ++j)
            dst[(size_t)j*QKVC + col0 + lane] = acc[j];
    }
}

__global__ __launch_bounds__(256)
void qkv_reduce(const float* __restrict__ qp,
                const float* __restrict__ bq, const float* __restrict__ bk,
                const float* __restrict__ bv, float* __restrict__ y)
{
    int idx = blockIdx.x*256 + threadIdx.x;      // over BS*QKVC
    if (idx >= BS*QKVC) return;
    int c = idx % QKVC;
    float s = (c < DIM) ? bq[c] : (c < DIM+1024 ? bk[c-DIM] : bv[c-DIM-1024]);
    for (int k = 0; k < KSPLIT; ++k) s += qp[(size_t)k*BS*QKVC + idx];
    y[idx] = s;
}

// ============================================================
// RoPE on q (cols 0..5119) and k (cols 5120..6143) in place.
// ============================================================
__global__ __launch_bounds__(256)
void rope_kernel(float* __restrict__ y, const float* __restrict__ fc)
{
    const int NQ = BS * NH * 64;
    const int NK = BS * NKV * 64;
    int idx = blockIdx.x * 256 + threadIdx.x;
    if (idx >= NQ + NK) return;
    size_t base; int i;
    if (idx < NQ) {
        int b = idx / (NH*64); int rem = idx % (NH*64);
        int h = rem / 64; i = rem % 64;
        base = (size_t)b*QKVC + h*HD + 2*i;
    } else {
        int j = idx - NQ;
        int b = j / (NKV*64); int rem = j % (NKV*64);
        int h = rem / 64; i = rem % 64;
        base = (size_t)b*QKVC + DIM + h*HD + 2*i;
    }
    float c = fc[2*i], s = fc[2*i + 1];
    float xe = y[base], xo = y[base + 1];
    y[base]     = xe*c - xo*s;
    y[base + 1] = xe*s + xo*c;
}

// ============================================================
// Flash-attention partial pass over one 256-key chunk.
// Scores via chained f32 WMMA; new token (key 4095) from workspace.
// ============================================================
__global__ __launch_bounds__(32)
void attn_partial(const float* __restrict__ y,
                  const float* __restrict__ cache_k,
                  const float* __restrict__ cache_v,
                  float* __restrict__ pm, float* __restrict__ pl,
                  float* __restrict__ po)
{
    __shared__ float ldsK[16 * 132];
    __shared__ float ldsP[NREP * 16];
    const int lane  = threadIdx.x;
    const int row   = lane & 15;
    const int offs  = (lane >> 4) << 1;
    const int chunk = blockIdx.x;
    const int g     = blockIdx.y;
    const int b     = blockIdx.z;

    const float* yb = y + (size_t)b * QKVC;

    // preload Q (5 reps, zero-padded to 16 rows) in WMMA A layout
    v2f qa[32];
    {
        const float* qrow = yb + (g*NREP + row) * HD;
#pragma unroll
        for (int t = 0; t < 32; ++t) {
            v2f v = {0.f, 0.f};
            if (row < NREP) v = *(const v2f*)&qrow[4*t + offs];
            qa[t] = v;
        }
    }

    float m[NREP], l[NREP];
    v4f o[NREP];
    for (int r = 0; r < NREP; ++r) { m[r] = -INFINITY; l[r] = 0.f; o[r] = (v4f){0,0,0,0}; }

    const float scale = 0.08838834764831845f;   // 1/sqrt(128)

    for (int tile = 0; tile < CH/16; ++tile) {
        const int kt0 = chunk*CH + tile*16;
        // stage 16 key rows into LDS (kt uniform across the wave)
        for (int rr = 0; rr < 16; ++rr) {
            const int kt = kt0 + rr;
            const float* kp = (kt == KLEN - 1)
                ? (yb + DIM + g*HD)
                : (cache_k + (((size_t)b*KLEN + kt)*NKV + g)*HD);
            __builtin_prefetch(kp + 16*NKV*HD, 0, 1);   // next tile, same row slot
            *(v4f*)&ldsK[rr*132 + lane*4] = *(const v4f*)&kp[lane*4];
        }
        __syncthreads();

        v8f s = {0.f,0.f,0.f,0.f,0.f,0.f,0.f,0.f};
#pragma unroll
        for (int t = 0; t < 32; ++t) {
            v2f kb = *(const v2f*)&ldsK[row*132 + 4*t + offs];
            s = __builtin_amdgcn_wmma_f32_16x16x4_f32(false, qa[t], false, kb,
                                                      (short)0, s, false, false);
        }
        __syncthreads();

        // online softmax (rows 0..4 live in VGPRs 0..4, lanes 0..15)
        float corr[NREP];
        for (int r = 0; r < NREP; ++r) {
            float sc = s[r] * scale;
            float tm = sc;
            tm = fmaxf(tm, __shfl_xor(tm, 8, 16));
            tm = fmaxf(tm, __shfl_xor(tm, 4, 16));
            tm = fmaxf(tm, __shfl_xor(tm, 2, 16));
            tm = fmaxf(tm, __shfl_xor(tm, 1, 16));
            tm = __shfl(tm, 0, 32);
            float mnew = fmaxf(m[r], tm);
            float p  = expf(sc - mnew);
            corr[r]  = expf(m[r] - mnew);
            float ps = p;
            ps += __shfl_xor(ps, 8, 16);
            ps += __shfl_xor(ps, 4, 16);
            ps += __shfl_xor(ps, 2, 16);
            ps += __shfl_xor(ps, 1, 16);
            ps = __shfl(ps, 0, 32);
            l[r] = l[r]*corr[r] + ps;
            m[r] = mnew;
            if (lane < 16) ldsP[r*16 + lane] = p;
            o[r] *= corr[r];
        }
        __syncthreads();

        // PV: lane owns d = lane*4..+3, V rows read coalesced (b128)
        for (int kk = 0; kk < 16; ++kk) {
            const int kt = kt0 + kk;
            const float* vp = (kt == KLEN - 1)
                ? (yb + DIM + NKV*HD + g*HD)
                : (cache_v + (((size_t)b*KLEN + kt)*NKV + g)*HD);
            v4f vv = *(const v4f*)&vp[lane*4];
            for (int r = 0; r < NREP; ++r)
                o[r] += vv * ldsP[r*16 + kk];
        }
        __syncthreads();
    }

    const int pidx = ((b*NKV + g)*NCHUNK + chunk) * NREP;
    if (lane == 0)
        for (int r = 0; r < NREP; ++r) { pm[pidx + r] = m[r]; pl[pidx + r] = l[r]; }
    for (int r = 0; r < NREP; ++r)
        *(v4f*)&po[(size_t)(pidx + r)*HD + lane*4] = o[r];
}

// ============================================================
// Merge chunk partials (log-sum-exp) -> att[b][h*128+d]
// ============================================================
__global__ __launch_bounds__(128)
void attn_reduce(const float* __restrict__ pm, const float* __restrict__ pl,
                 const float* __restrict__ po, float* __restrict__ att)
{
    const int d   = threadIdx.x;
    const int bgr = blockIdx.x;
    const int b   = bgr / (NKV*NREP);
    const int rem = bgr % (NKV*NREP);
    const int g   = rem / NREP;
    const int r   = rem % NREP;

    float M = -INFINITY;
    for (int c = 0; c < NCHUNK; ++c)
        M = fmaxf(M, pm[((b*NKV + g)*NCHUNK + c)*NREP + r]);
    float L = 0.f, acc = 0.f;
    for (int c = 0; c < NCHUNK; ++c) {
        int idx = ((b*NKV + g)*NCHUNK + c)*NREP + r;
        float w = expf(pm[idx] - M);
        L   += pl[idx] * w;
        acc += po[(size_t)idx*HD + d] * w;
    }
    att[(size_t)b*DIM + (g*NREP + r)*HD + d] = acc / L;
}

// ============================================================
// Output projection partials: op[ks][b][0:5119]
// ============================================================
__global__ __launch_bounds__(32)
void out_gemv(const float* __restrict__ att, const float* __restrict__ wo,
              float* __restrict__ op)
{
    __shared__ float ldsX[16 * 132];
    const int lane = threadIdx.x;
    for (int j = lane; j < 8*132; j += 32) ldsX[8*132 + j] = 0.f;
    __syncthreads();

    const int col0 = blockIdx.x * 16;
    const int ks   = blockIdx.y;
    const int dt0  = ks * (DIM/128/KSPLIT);
    const int dt1  = dt0 + (DIM/128/KSPLIT);

    v8f acc = {0.f,0.f,0.f,0.f,0.f,0.f,0.f,0.f};
    gemv_ktile<DIM>(wo, att, col0, dt0, dt1, lane, acc, ldsX);

    if (lane < 16) {
        float* dst = op + (size_t)ks * BS * DIM;
        for (int j = 0; j < 8; ++j)
            dst[(size_t)j*DIM + col0 + lane] = acc[j];
    }
}

__global__ __launch_bounds__(256)
void out_reduce(const float* __restrict__ op, const float* __restrict__ bo,
                float* __restrict__ out)
{
    int idx = blockIdx.x*256 + threadIdx.x;      // over BS*DIM
    if (idx >= BS*DIM) return;
    int c = idx % DIM;
    float s = bo[c];
    for (int k = 0; k < KSPLIT; ++k) s += op[(size_t)k*BS*DIM + idx];
    out[idx] = s;
}

// ============================================================
extern "C" void kernel_launch(void* const* d_in, const int* in_sizes, int n_in,
                              void* d_out, int out_size, void* d_ws, size_t ws_size,
                              hipStream_t stream)
{
    const float* x  = (const float*)d_in[0];
    // d_in[1] = start_pos (compile-time constant 4095 here)
    const float* fc = (const float*)d_in[2];
    const float* ck = (const float*)d_in[3];
    const float* cv = (const float*)d_in[4];
    const float* wq = (const float*)d_in[5];
    const float* bq = (const float*)d_in[6];
    const float* wk = (const float*)d_in[7];
    const float* bk = (const float*)d_in[8];
    const float* wv = (const float*)d_in[9];
    const float* bv = (const float*)d_in[10];
    const float* wo = (const float*)d_in[11];
    const float* bo = (const float*)d_in[12];
    float* out = (float*)d_out;
    float* ws  = (float*)d_ws;

    float* y   = ws + WS_Y;
    float* att = ws + WS_ATT;
    float* pm  = ws + WS_PM;
    float* pl  = ws + WS_PL;
    float* po  = ws + WS_PO;
    float* qp  = ws + WS_QP;
    float* op  = ws + WS_OP;

    qkv_gemv<<<dim3(QKVC/16, KSPLIT), 32, 0, stream>>>(x, wq, wk, wv, qp);
    qkv_reduce<<<(BS*QKVC + 255)/256, 256, 0, stream>>>(qp, bq, bk, bv, y);
    rope_kernel<<<(BS*NH*64 + BS*NKV*64 + 255)/256, 256, 0, stream>>>(y, fc);
    attn_partial<<<dim3(NCHUNK, NKV, BS), 32, 0, stream>>>(y, ck, cv, pm, pl, po);
    attn_reduce<<<BS*NKV*NREP, 128, 0, stream>>>(pm, pl, po, att);
    out_gemv<<<dim3(DIM/16, KSPLIT), 32, 0, stream>>>(att, wo, op);
    out_reduce<<<(BS*DIM + 255)/256, 256, 0, stream>>>(op, bo, out);
}